// QuatRgsLoss_63591285785229
// MI455X (gfx1250) — compile-verified
//
#include <hip/hip_runtime.h>
#include <hip/hip_bf16.h>
#include <math.h>

typedef __attribute__((ext_vector_type(2))) float v2f;
typedef __attribute__((ext_vector_type(8))) float v8f;

#define NPTS 2048
#define NBATCH 16
#define NTILES (NPTS / 16)   // 128 tiles of 16 along each of n and m

__device__ __forceinline__ void quat_to_mat_dev(const float q_in[4], float R[9]) {
    float w = q_in[0], x = q_in[1], y = q_in[2], z = q_in[3];
    float nrm = sqrtf(w * w + x * x + y * y + z * z);
    nrm = fmaxf(nrm, 1e-8f);
    float inv = 1.0f / nrm;
    w *= inv; x *= inv; y *= inv; z *= inv;
    R[0] = 1.0f - 2.0f * (y * y + z * z);
    R[1] = 2.0f * (x * y - w * z);
    R[2] = 2.0f * (x * z + w * y);
    R[3] = 2.0f * (x * y + w * z);
    R[4] = 1.0f - 2.0f * (x * x + z * z);
    R[5] = 2.0f * (y * z - w * x);
    R[6] = 2.0f * (x * z - w * y);
    R[7] = 2.0f * (y * z + w * x);
    R[8] = 1.0f - 2.0f * (x * x + y * y);
}

// Kernel 1: rotate points and emit WMMA fragment-ready tiles.
// For each 16-wide tile t and lane L (wave32 V_WMMA_F32_16X16X4_F32 layout):
//   A-frag (gt, pre-scaled by -2, K=3 padded to 4):
//     L<16 : (-2*g0[n0+L], -2*g1[n0+L])   (K=0,1)
//     L>=16: (-2*g2[n0+L-16], 0)          (K=2,3)
//   B-frag (pred): same shape without the -2 scale.
// Also xsq = |g|^2 (unscaled), ysq = |q|^2.
__global__ __launch_bounds__(256) void rotate_pack_kernel(
    const float* __restrict__ predquat, const float* __restrict__ gtquat,
    const float* __restrict__ points,
    v2f* __restrict__ afrag, v2f* __restrict__ bfrag,
    float* __restrict__ xsq, float* __restrict__ ysq) {
    const int b = blockIdx.x;
    float pq[4], gq[4];
#pragma unroll
    for (int i = 0; i < 4; ++i) { pq[i] = predquat[b * 4 + i]; gq[i] = gtquat[b * 4 + i]; }
    float Rp[9], Rg[9];
    quat_to_mat_dev(pq, Rp);
    quat_to_mat_dev(gq, Rg);

    const float* P = points + (size_t)b * 3 * NPTS;

    for (int n = threadIdx.x; n < NPTS; n += blockDim.x) {
        float p0 = P[n], p1 = P[NPTS + n], p2 = P[2 * NPTS + n];
        float g0 = Rg[0] * p0 + Rg[1] * p1 + Rg[2] * p2;
        float g1 = Rg[3] * p0 + Rg[4] * p1 + Rg[5] * p2;
        float g2 = Rg[6] * p0 + Rg[7] * p1 + Rg[8] * p2;
        float q0 = Rp[0] * p0 + Rp[1] * p1 + Rp[2] * p2;
        float q1 = Rp[3] * p0 + Rp[4] * p1 + Rp[5] * p2;
        float q2 = Rp[6] * p0 + Rp[7] * p1 + Rp[8] * p2;

        const int tile = n >> 4;
        const int l16  = n & 15;
        const size_t base = ((size_t)(b * NTILES + tile)) * 32;

        v2f alo; alo.x = -2.0f * g0; alo.y = -2.0f * g1;
        v2f ahi; ahi.x = -2.0f * g2; ahi.y = 0.0f;
        afrag[base + l16]      = alo;
        afrag[base + 16 + l16] = ahi;

        v2f blo; blo.x = q0; blo.y = q1;
        v2f bhi; bhi.x = q2; bhi.y = 0.0f;
        bfrag[base + l16]      = blo;
        bfrag[base + 16 + l16] = bhi;

        xsq[b * NPTS + n] = g0 * g0 + g1 * g1 + g2 * g2;
        ysq[b * NPTS + n] = q0 * q0 + q1 * q1 + q2 * q2;
    }
}

// Kernel 2: fused cross-matmul + row-min via V_WMMA_F32_16X16X4_F32.
// One wave per 16-column m-tile; loops over all 128 n-tiles.
// D = A*B + C with A = -2*gt tile, B = pred tile, C[i][j] = xsq[n0+i]
//   => D[i][j] = xsq[n_i] - 2*cross[n_i, m_j].
// Hot loop: 1x global_load_b64 + 2x ds_load_b128 + 1x wmma + 4x v_min3.
__global__ __launch_bounds__(256) void wmma_min_kernel(
    const v2f* __restrict__ afrag, const v2f* __restrict__ bfrag,
    const float* __restrict__ xsq, const float* __restrict__ ysq,
    float* __restrict__ partial) {
    __shared__ float s_xsq[NPTS];
    const int b   = blockIdx.y;
    const int tid = threadIdx.x;

    for (int i = tid; i < NPTS; i += 256) s_xsq[i] = xsq[b * NPTS + i];
    __syncthreads();

    const int wave   = tid >> 5;        // 0..7
    const int lane   = tid & 31;
    const int half   = lane >> 4;
    const int lane16 = lane & 15;
    const int mtile  = blockIdx.x * 8 + wave;   // 0..127
    const int halfoff = half * 8;

    const v2f* Afr = afrag + ((size_t)b * NTILES) * 32;
    const v2f  bf  = bfrag[((size_t)(b * NTILES + mtile)) * 32 + lane];

    float runmin = 3.0e38f;
    for (int nt = 0; nt < NTILES; ++nt) {
        const v2f af = Afr[nt * 32 + lane];              // single b64 load, no branches
        v8f c;
        const int nbase = nt * 16 + halfoff;
#pragma unroll
        for (int r = 0; r < 8; ++r) c[r] = s_xsq[nbase + r];

        v8f d = __builtin_amdgcn_wmma_f32_16x16x4_f32(
            /*neg_a=*/false, af, /*neg_b=*/false, bf,
            /*c_mod=*/(short)0, c, /*reuse_a=*/false, /*reuse_b=*/false);

#pragma unroll
        for (int r = 0; r < 8; ++r) runmin = fminf(runmin, d[r]);
    }
    // combine the two 8-row halves for each column
    runmin = fminf(runmin, __shfl_xor(runmin, 16, 32));
    // add the per-column ysq term
    float m_l2 = runmin + ysq[b * NPTS + mtile * 16 + lane16];
    // sum the 16 columns of this tile (fixed shuffle tree -> deterministic)
    float s = m_l2;
    s += __shfl_xor(s, 1, 32);
    s += __shfl_xor(s, 2, 32);
    s += __shfl_xor(s, 4, 32);
    s += __shfl_xor(s, 8, 32);
    if (lane == 0) partial[b * NTILES + mtile] = s;
}

// Kernel 3: deterministic final reduction + cos loss.
__global__ __launch_bounds__(256) void final_kernel(
    const float* __restrict__ predquat, const float* __restrict__ gtquat,
    const float* __restrict__ partial, float* __restrict__ out) {
    __shared__ float sdata[256];
    const int tid = threadIdx.x;
    float s = 0.0f;
    for (int i = tid; i < NBATCH * NTILES; i += 256) s += partial[i];
    sdata[tid] = s;
    __syncthreads();
    for (int off = 128; off > 0; off >>= 1) {
        if (tid < off) sdata[tid] += sdata[tid + off];
        __syncthreads();
    }
    if (tid == 0) {
        float pt = sdata[0] / ((float)NBATCH * (float)NPTS);
        if (isnan(pt)) pt = 0.0f;
        float acc = 0.0f;
        for (int b = 0; b < NBATCH; ++b) {
            float dot = 0.0f, np = 0.0f, ng = 0.0f;
#pragma unroll
            for (int i = 0; i < 4; ++i) {
                float p = predquat[b * 4 + i], g = gtquat[b * 4 + i];
                dot += p * g; np += p * p; ng += g * g;
            }
            float denom = fmaxf(sqrtf(np) * sqrtf(ng), 1e-8f);
            acc += 1.0f - dot / denom;
        }
        out[0] = acc / (float)NBATCH;
        out[1] = pt;
    }
}

extern "C" void kernel_launch(void* const* d_in, const int* in_sizes, int n_in,
                              void* d_out, int out_size, void* d_ws, size_t ws_size,
                              hipStream_t stream) {
    const float* predquat = (const float*)d_in[0];  // (16,4)
    const float* gtquat   = (const float*)d_in[1];  // (16,4)
    const float* points   = (const float*)d_in[2];  // (16,3,2048)
    float* out = (float*)d_out;                     // [cos_loss, pt_loss]

    float* ws = (float*)d_ws;
    // fragment buffers: NBATCH * NTILES tiles * 32 lanes * float2
    v2f*   afrag = (v2f*)ws;                                        // 131072 floats
    v2f*   bfrag = (v2f*)(ws + (size_t)NBATCH * NTILES * 64);       // 131072 floats
    float* xsq   = ws + (size_t)2 * NBATCH * NTILES * 64;           // 32768 floats
    float* ysq   = xsq + (size_t)NBATCH * NPTS;                     // 32768 floats
    float* partial = ysq + (size_t)NBATCH * NPTS;                   // 2048 floats

    rotate_pack_kernel<<<NBATCH, 256, 0, stream>>>(predquat, gtquat, points,
                                                   afrag, bfrag, xsq, ysq);
    wmma_min_kernel<<<dim3(NTILES / 8, NBATCH), 256, 0, stream>>>(
        afrag, bfrag, xsq, ysq, partial);
    final_kernel<<<1, 256, 0, stream>>>(predquat, gtquat, partial, out);
}